// ABS_LSTM5_4_62586263437988
// MI455X (gfx1250) — compile-verified
//
#include <hip/hip_runtime.h>
#include <hip/hip_bf16.h>
#include <math.h>

// ---------------------------------------------------------------------------
// ABS-LSTM for MI455X (gfx1250, wave32, WMMA).
// - All GEMMs use V_WMMA_F32_16X16X4_F32 (exact f32, matches f32 reference).
// - Weights pre-packed into per-lane fragment layout -> B frag = 1x b64 load.
// - Global->LDS tile staging uses async-to-LDS (ASYNCcnt) path.
// ---------------------------------------------------------------------------

typedef __attribute__((ext_vector_type(2))) float v2f;
typedef __attribute__((ext_vector_type(8))) float v8f;

#define Bn 128
#define Tn 512
#define Fn 512
#define Un 256

__device__ __forceinline__ v8f wmma4(v2f a, v2f b, v8f c) {
  return __builtin_amdgcn_wmma_f32_16x16x4_f32(false, a, false, b, (short)0, c,
                                               false, false);
}

__device__ __forceinline__ float hsig(float x) {
  return fminf(fmaxf(0.2f * x + 0.5f, 0.0f), 1.0f);
}

__device__ __forceinline__ float rowsum16(float v) {
  v += __shfl_xor(v, 1, 32);
  v += __shfl_xor(v, 2, 32);
  v += __shfl_xor(v, 4, 32);
  v += __shfl_xor(v, 8, 32);
  return v;
}

// CDNA5 async global->LDS copy (tracked by ASYNCcnt).
__device__ __forceinline__ void async_copy_b128(unsigned lds_byte_addr,
                                                const float* g) {
  asm volatile("global_load_async_to_lds_b128 %0, %1, off" ::"v"(lds_byte_addr),
               "v"(g)
               : "memory");
}
__device__ __forceinline__ void wait_async0() {
  asm volatile("s_wait_asynccnt 0x0" ::: "memory");
}

// ---------------------------------------------------------------------------
// Weight pre-pack: W (K x N, row-major) -> P[ntile][k/4][lane][2] so that a
// wave's WMMA B-fragment is one contiguous b64 per lane (256B per wave/frag).
// Fragment def: lane = hi*16+lr reads W[k+2*hi][n0+lr], W[k+2*hi+1][n0+lr].
// ---------------------------------------------------------------------------
__global__ void pack_weight(const float* __restrict__ W, float* __restrict__ P,
                            int K, int N) {
  const int idx = blockIdx.x * blockDim.x + threadIdx.x;
  const int K4 = K >> 2;
  const int total = (N >> 4) * K4 * 32;
  if (idx >= total) return;
  const int lane = idx & 31;
  const int kk = (idx >> 5) % K4;
  const int nt = (idx >> 5) / K4;
  const int hi = lane >> 4, lr = lane & 15;
  const int r0 = kk * 4 + 2 * hi;
  P[(size_t)idx * 2 + 0] = W[(size_t)r0 * N + nt * 16 + lr];
  P[(size_t)idx * 2 + 1] = W[((size_t)r0 + 1) * N + nt * 16 + lr];
}

// ---------------------------------------------------------------------------
// Kernel A: precompute  xg = x_tbf @ kernel (T,B,4U)
//                       xh = x_tbf @ Wh_x   (T,B,U)
//                       xc = x_tbf @ Wc_x   (T,B,U)
// grid = (T*B/16, 12), block = 256 (8 waves, 1 col-tile each)
// ---------------------------------------------------------------------------
#define LDA_A 516  // 512 + 4 pad (516 % 64 == 4 -> bank-conflict-free)

__global__ void abs_lstm_pre(const float* __restrict__ x,
                             const float* __restrict__ Pk,
                             const float* __restrict__ PhX,
                             const float* __restrict__ PcX,
                             float* __restrict__ xg, float* __restrict__ xh,
                             float* __restrict__ xc) {
  extern __shared__ float lds[];  // 16 x LDA_A
  const int rt = blockIdx.x;
  const int t = rt >> 3;
  const int b0 = (rt & 7) * 16;

  // async-stage A tile (16 rows x 512) : 2048 x b128
  for (int i = threadIdx.x; i < 16 * 128; i += blockDim.x) {
    const int r = i >> 7;
    const int k4 = (i & 127) * 4;
    async_copy_b128((unsigned)(uintptr_t)&lds[r * LDA_A + k4],
                    &x[((size_t)(b0 + r) * Tn + t) * Fn + k4]);
  }
  wait_async0();
  __syncthreads();

  const int wave = threadIdx.x >> 5;
  const int lane = threadIdx.x & 31;
  const int lr = lane & 15, hi = lane >> 4;
  const int ct = blockIdx.y * 8 + wave;  // 0..95 column tiles

  const v2f* Bp;
  float* dst;
  int n0, ldd;
  if (ct < 64) {
    Bp = (const v2f*)Pk + ((size_t)ct * 128) * 32 + lane;
    n0 = ct * 16;        dst = xg; ldd = 4 * Un;
  } else if (ct < 80) {
    Bp = (const v2f*)PhX + ((size_t)(ct - 64) * 128) * 32 + lane;
    n0 = (ct - 64) * 16; dst = xh; ldd = Un;
  } else {
    Bp = (const v2f*)PcX + ((size_t)(ct - 80) * 128) * 32 + lane;
    n0 = (ct - 80) * 16; dst = xc; ldd = Un;
  }

  v8f c = {};
  for (int kk = 0; kk < 128; ++kk) {
    const v2f a = *(const v2f*)&lds[lr * LDA_A + kk * 4 + 2 * hi];
    c = wmma4(a, Bp[kk * 32], c);
  }

  const size_t rowbase = (size_t)t * Bn + b0;
#pragma unroll
  for (int i = 0; i < 8; i++)
    dst[(rowbase + i + 8 * hi) * (size_t)ldd + n0 + lr] = c[i];
}

// ---------------------------------------------------------------------------
// Kernel B: sequential scan. grid = 16 (2 dirs x 8 batch tiles of 16),
// block = 512 (16 waves). Recurrent state in LDS, slots rotated.
// ---------------------------------------------------------------------------
#define LDP 260  // 256 + 4 pad
#define TILEF (16 * LDP)
#define OFF_H 0
#define OFF_C (4 * TILEF)
#define OFF_HMIX (8 * TILEF)
#define OFF_CMIX (9 * TILEF)
#define OFF_SCORE (10 * TILEF)     // [2][4][16]
#define OFF_WTS (OFF_SCORE + 128)  // [2][16][4]
#define SCAN_LDS_FLOATS (OFF_WTS + 128)

__global__ void abs_lstm_scan(const float* __restrict__ xg,
                              const float* __restrict__ xh,
                              const float* __restrict__ xc,
                              const float* __restrict__ PhS,
                              const float* __restrict__ PcS,
                              const float* __restrict__ Prk,
                              float* __restrict__ outs) {
  extern __shared__ float lds[];
  const int dir = blockIdx.x >> 3;
  const int b0 = (blockIdx.x & 7) * 16;

  const int wave = threadIdx.x >> 5;
  const int lane = threadIdx.x & 31;
  const int lr = lane & 15, hi = lane >> 4;
  const int n0 = wave * 16;

  // packed B bases for this wave's column tile
  const v2f* BhS = (const v2f*)PhS + ((size_t)wave * 64) * 32 + lane;
  const v2f* BcS = (const v2f*)PcS + ((size_t)wave * 64) * 32 + lane;
  const v2f* Brk[4];
#pragma unroll
  for (int g = 0; g < 4; g++)
    Brk[g] = (const v2f*)Prk + ((size_t)(g * 16 + wave) * 64) * 32 + lane;

  for (int i = threadIdx.x; i < 8 * TILEF; i += blockDim.x) lds[i] = 0.0f;
  __syncthreads();

  int rot = 0;
  for (int step = 0; step < Tn; ++step) {
    const int t = dir ? (Tn - 1 - step) : step;

    if (threadIdx.x < 128) lds[OFF_SCORE + threadIdx.x] = 0.0f;
    __syncthreads();

    // ---- phase 2: 8 attention GEMMs  a = tanh(state @ Ws + xpart) ----
    for (int combo = 0; combo < 8; ++combo) {
      const int type = combo >> 2;  // 0 = h-att, 1 = c-att
      const int st = combo & 3;
      const int slot = (rot + st) & 3;
      const float* A = &lds[(type ? OFF_C : OFF_H) + slot * TILEF];
      // reference quirk: c-attention state 4 uses Wh (and Wh's x-part)
      const v2f* Bp = (type == 0) ? BhS : (st < 3 ? BcS : BhS);
      const float* xp = (type == 0) ? xh : (st < 3 ? xc : xh);

      v8f c = {};
      for (int kk = 0; kk < 64; ++kk) {
        const v2f a = *(const v2f*)&A[lr * LDP + kk * 4 + 2 * hi];
        c = wmma4(a, Bp[kk * 32], c);
      }
#pragma unroll
      for (int i = 0; i < 8; i++) {
        const int r = i + 8 * hi;
        float v = c[i] + xp[((size_t)t * Bn + b0 + r) * Un + n0 + lr];
        v = tanhf(v);
        float sq = rowsum16(v * v);
        if (lr == 0) atomicAdd(&lds[OFF_SCORE + (type * 4 + st) * 16 + r], sq);
      }
    }
    __syncthreads();

    // ---- phase 3: sqrt + softmax over 4 states, per row ----
    if (threadIdx.x < 32) {
      const int type = threadIdx.x >> 4, r = threadIdx.x & 15;
      float s[4], mx = -1e30f;
#pragma unroll
      for (int j = 0; j < 4; j++) {
        s[j] = sqrtf(lds[OFF_SCORE + (type * 4 + j) * 16 + r]);
        mx = fmaxf(mx, s[j]);
      }
      float den = 0.0f;
#pragma unroll
      for (int j = 0; j < 4; j++) {
        s[j] = __expf(s[j] - mx);
        den += s[j];
      }
#pragma unroll
      for (int j = 0; j < 4; j++)
        lds[OFF_WTS + type * 64 + r * 4 + j] = s[j] / den;
    }
    __syncthreads();

    // ---- phase 4: hmix / cmix = relu(sum_j w_j * state_j) ----
    for (int i = threadIdx.x; i < 2 * 16 * Un; i += blockDim.x) {
      const int type = i >> 12;
      const int r = (i >> 8) & 15;
      const int u = i & 255;
      const int base = type ? OFF_C : OFF_H;
      float acc = 0.0f;
#pragma unroll
      for (int j = 0; j < 4; j++)
        acc += lds[OFF_WTS + type * 64 + r * 4 + j] *
               lds[base + ((rot + j) & 3) * TILEF + r * LDP + u];
      lds[(type ? OFF_CMIX : OFF_HMIX) + r * LDP + u] = fmaxf(acc, 0.0f);
    }
    __syncthreads();

    // ---- phase 5: g = xg_t + hmix @ rk ; gates ; state update ----
    v8f ci = {}, cf = {}, cm = {}, co = {};
    for (int kk = 0; kk < 64; ++kk) {
      const v2f a = *(const v2f*)&lds[OFF_HMIX + lr * LDP + kk * 4 + 2 * hi];
      ci = wmma4(a, Brk[0][kk * 32], ci);
      cf = wmma4(a, Brk[1][kk * 32], cf);
      cm = wmma4(a, Brk[2][kk * 32], cm);
      co = wmma4(a, Brk[3][kk * 32], co);
    }

    const int newSlot = (rot + 3) & 3;  // overwrite the oldest (h4/c4)
#pragma unroll
    for (int i = 0; i < 8; i++) {
      const int r = i + 8 * hi;
      const int u = n0 + lr;
      const size_t gb = ((size_t)t * Bn + b0 + r) * (4 * Un);
      const float gi = hsig(ci[i] + xg[gb + u]);
      const float gf = hsig(cf[i] + xg[gb + Un + u]);
      const float gm = hsig(cm[i] + xg[gb + 2 * Un + u]);
      const float go = hsig(co[i] + xg[gb + 3 * Un + u]);
      const float cn = gf * lds[OFF_CMIX + r * LDP + u] + gi * gm;
      const float hn = go * tanhf(cn);
      lds[OFF_H + newSlot * TILEF + r * LDP + u] = hn;
      lds[OFF_C + newSlot * TILEF + r * LDP + u] = cn;
      outs[((size_t)(b0 + r) * Tn + t) * (2 * Un) + dir * Un + u] = hn;
    }
    rot = newSlot;
    __syncthreads();
  }
}

// ---------------------------------------------------------------------------
// Kernel C: final attention. grid = B*T/16, block = 512 (16 waves).
// ---------------------------------------------------------------------------
#define LDC 516
#define OFF_O 0
#define OFF_X (16 * LDC)
#define OFF_HA (2 * 16 * LDC)
#define OFF_HB (OFF_HA + 16)
#define FIN_LDS_FLOATS (OFF_HB + 16)

__global__ void abs_lstm_final(const float* __restrict__ outs,
                               const float* __restrict__ x,
                               const float* __restrict__ PA11,
                               const float* __restrict__ PA12,
                               const float* __restrict__ a2,
                               const float* __restrict__ a2_2,
                               float* __restrict__ out) {
  extern __shared__ float lds[];
  const size_t row0 = (size_t)blockIdx.x * 16;

  // async-stage both 16x512 tiles : 4096 x b128
  for (int i = threadIdx.x; i < 2 * 16 * 128; i += blockDim.x) {
    const int half = i >> 11;
    const int r = (i >> 7) & 15;
    const int k4 = (i & 127) * 4;
    const float* src = half ? &x[(row0 + r) * 512 + k4]
                            : &outs[(row0 + r) * 512 + k4];
    const int dsto = half ? OFF_X : OFF_O;
    async_copy_b128((unsigned)(uintptr_t)&lds[dsto + r * LDC + k4], src);
  }
  if (threadIdx.x < 32) lds[OFF_HA + threadIdx.x] = 0.0f;
  wait_async0();
  __syncthreads();

  const int wave = threadIdx.x >> 5;
  const int lane = threadIdx.x & 31;
  const int lr = lane & 15, hi = lane >> 4;
  const int n0 = wave * 16;

#pragma unroll
  for (int half = 0; half < 2; ++half) {
    const float* Alds = half ? &lds[OFF_X] : &lds[OFF_O];
    const v2f* Bp = (const v2f*)(half ? PA12 : PA11) +
                    ((size_t)wave * 128) * 32 + lane;
    const float* vec = half ? a2_2 : a2;
    const int dsto = half ? OFF_HB : OFF_HA;

    v8f c = {};
    for (int kk = 0; kk < 128; ++kk) {
      const v2f a = *(const v2f*)&Alds[lr * LDC + kk * 4 + 2 * hi];
      c = wmma4(a, Bp[kk * 32], c);
    }
    const float av = vec[n0 + lr];
#pragma unroll
    for (int i = 0; i < 8; i++) {
      const float v = rowsum16(fmaxf(c[i], 0.0f) * av);
      if (lr == 0) atomicAdd(&lds[dsto + i + 8 * hi], v);
    }
  }
  __syncthreads();

  for (int i = threadIdx.x; i < 16 * 512; i += blockDim.x) {
    const int r = i >> 9, u = i & 511;
    out[(row0 + r) * 512 + u] =
        tanhf(lds[OFF_HA + r] * lds[OFF_O + r * LDC + u] + lds[OFF_HB + r]);
  }
}

// ---------------------------------------------------------------------------
extern "C" void kernel_launch(void* const* d_in, const int* in_sizes, int n_in,
                              void* d_out, int out_size, void* d_ws,
                              size_t ws_size, hipStream_t stream) {
  const float* x     = (const float*)d_in[0];  // (B,T,F)
  const float* kern  = (const float*)d_in[1];  // (F,4U)
  const float* rk    = (const float*)d_in[2];  // (U,4U)
  const float* attnH = (const float*)d_in[3];  // (F+U,U)
  const float* attnC = (const float*)d_in[4];  // (F+U,U)
  const float* attn1 = (const float*)d_in[5];  // (2U+F,U)
  const float* a2    = (const float*)d_in[7];  // (U,1)
  const float* a2_2  = (const float*)d_in[8];  // (U,1)
  float* out = (float*)d_out;                  // (B,T,2U)

  // workspace partition (floats): intermediates + packed weights (~518 MB)
  float* p = (float*)d_ws;
  float* xg   = p; p += (size_t)Tn * Bn * 4 * Un;
  float* xh   = p; p += (size_t)Tn * Bn * Un;
  float* xc   = p; p += (size_t)Tn * Bn * Un;
  float* outs = p; p += (size_t)Bn * Tn * 2 * Un;
  float* Pk   = p; p += (size_t)Fn * 4 * Un;  // kernel (512 x 1024)
  float* PhX  = p; p += (size_t)Fn * Un;      // attn_h rows U.. (512 x 256)
  float* PcX  = p; p += (size_t)Fn * Un;
  float* PhS  = p; p += (size_t)Un * Un;      // attn_h rows 0..U-1 (256 x 256)
  float* PcS  = p; p += (size_t)Un * Un;
  float* Prk  = p; p += (size_t)Un * 4 * Un;  // rk (256 x 1024)
  float* PA11 = p; p += (size_t)(2 * Un) * Un;
  float* PA12 = p; p += (size_t)Fn * Un;

  // 0: pack all weights into WMMA fragment layout (one-time, ~5.8 MB)
  auto pack = [&](const float* W, float* P, int K, int N) {
    const int total = (N >> 4) * (K >> 2) * 32;
    pack_weight<<<(total + 255) / 256, 256, 0, stream>>>(W, P, K, N);
  };
  pack(kern, Pk, Fn, 4 * Un);
  pack(attnH + Un * Un, PhX, Fn, Un);
  pack(attnC + Un * Un, PcX, Fn, Un);
  pack(attnH, PhS, Un, Un);
  pack(attnC, PcS, Un, Un);
  pack(rk, Prk, Un, 4 * Un);
  pack(attn1, PA11, 2 * Un, Un);
  pack(attn1 + 2 * Un * Un, PA12, Fn, Un);

  // A: fused precompute GEMMs
  abs_lstm_pre<<<dim3((Tn * Bn) / 16, 12), 256, 16 * LDA_A * 4, stream>>>(
      x, Pk, PhX, PcX, xg, xh, xc);

  // B: persistent recurrent scan (2 dirs x 8 batch tiles)
  abs_lstm_scan<<<16, 512, SCAN_LDS_FLOATS * 4, stream>>>(xg, xh, xc, PhS, PcS,
                                                          Prk, outs);

  // C: final attention + tanh
  abs_lstm_final<<<(Bn * Tn) / 16, 512, FIN_LDS_FLOATS * 4, stream>>>(
      outs, x, PA11, PA12, a2, a2_2, out);
}